// EdgeAttention_57234734186660
// MI455X (gfx1250) — compile-verified
//
#include <hip/hip_runtime.h>
#include <math.h>

typedef __attribute__((ext_vector_type(2))) float v2f;
typedef __attribute__((ext_vector_type(8))) float v8f;

// ---------- order-preserving float <-> uint encoding (for atomic segment max) ----------
__device__ __forceinline__ unsigned enc_f(float f) {
    unsigned u = __float_as_uint(f);
    return (u & 0x80000000u) ? ~u : (u | 0x80000000u);
}
__device__ __forceinline__ float dec_f(unsigned u) {
    return __uint_as_float((u & 0x80000000u) ? (u ^ 0x80000000u) : ~u);
}
#define ENC_NEG_INF 0x007FFFFFu  // enc(-inf)

// ---------- kernel 0: weight transposes + segment-state init ----------
__global__ void prep_weights(const float* __restrict__ Wi, const float* __restrict__ Wj,
                             const float* __restrict__ We,
                             float* __restrict__ Wcat,   // [C][2H]  row-major
                             float* __restrict__ WeT,    // [De][H]  row-major
                             int C, int H, int De) {
    int idx = blockIdx.x * blockDim.x + threadIdx.x;
    int HO = 2 * H;
    if (idx < C * HO) {
        int k = idx / HO;
        int n = idx - k * HO;
        Wcat[idx] = (n < H) ? Wi[(size_t)n * C + k] : Wj[(size_t)(n - H) * C + k];
    }
    if (idx < De * H) {
        int k = idx / H;
        int n = idx - k * H;
        WeT[idx] = We[(size_t)n * De + k];
    }
}

__global__ void init_seg(unsigned* __restrict__ seg_max_u, float* __restrict__ seg_sum, int N) {
    int i = blockIdx.x * blockDim.x + threadIdx.x;
    if (i < N) { seg_max_u[i] = ENC_NEG_INF; seg_sum[i] = 0.0f; }
}

// ---------- kernel 1: hcat[n][0:64] = x@Wi^T , hcat[n][64:128] = x@Wj^T ----------
// One wave per 16x16 tile of the [N x 128] output. fp32 WMMA, K stepped by 4.
__global__ __launch_bounds__(256) void node_proj(const float* __restrict__ x,
                                                 const float* __restrict__ Wcat,
                                                 float* __restrict__ hcat,
                                                 int Nn, int C, int HO) {
    const int lane   = threadIdx.x & 31;
    const int wave   = threadIdx.x >> 5;      // 8 waves -> 8 n-tiles (HO = 128)
    const int m_base = blockIdx.x * 16;
    const int n_base = wave * 16;
    const int lm     = lane & 15;
    const int koff   = (lane >> 4) * 2;       // lanes 0-15: K+0,K+1 ; lanes 16-31: K+2,K+3
    if (m_base >= Nn) return;                 // wave-uniform

    const float2* xrow = (const float2*)(x + (size_t)(m_base + lm) * C);
    const float*  bcol = Wcat + n_base + lm;  // column base; rows stride HO
    v8f c = {};
    #pragma unroll 8
    for (int k0 = 0; k0 < C; k0 += 4) {
        float2 af = xrow[(k0 + koff) >> 1];   // 8B contiguous load per lane
        v2f a; a.x = af.x; a.y = af.y;
        v2f b;
        b.x = bcol[(size_t)(k0 + koff)     * HO];
        b.y = bcol[(size_t)(k0 + koff + 1) * HO];
        c = __builtin_amdgcn_wmma_f32_16x16x4_f32(false, a, false, b, (short)0, c,
                                                  false, false);
    }
    // D layout: VGPR v, lanes 0-15 -> M=v ; lanes 16-31 -> M=v+8 ; N=lane&15
    const int halfm = (lane >> 4) * 8;
    float* out = hcat + (size_t)m_base * HO + n_base + lm;
    #pragma unroll
    for (int v = 0; v < 8; ++v) {
        int m = halfm + v;
        if (m_base + m < Nn) out[(size_t)m * HO] = c[v];
    }
}

// ---------- kernel 2: per-16-edge tile: he GEMM + gathers + tanh + score + seg max ----------
// W_e^T (16 KB) + w_score staged in LDS (320 KB/WGP on CDNA5); B operand read via ds_load.
__global__ __launch_bounds__(256) void edge_logits(const float* __restrict__ edge_attr,
                                                   const float* __restrict__ WeT,
                                                   const float* __restrict__ hcat,
                                                   const int* __restrict__ eidx, // [2][E]: row0=j(src), row1=i(dst)
                                                   const float* __restrict__ w_score,
                                                   float* __restrict__ logits,
                                                   unsigned* __restrict__ seg_max_u,
                                                   int E, int De, int H, int HO) {
    __shared__ float sWeT[64 * 64];           // [K=De][N=H], De*H <= 4096
    __shared__ float sWsc[64];
    for (int idx = threadIdx.x; idx < De * H; idx += 256) sWeT[idx] = WeT[idx];
    if ((int)threadIdx.x < H) sWsc[threadIdx.x] = w_score[threadIdx.x];
    __syncthreads();                          // all threads reach this (guard comes after)

    const int  lane   = threadIdx.x & 31;
    const int  wave   = threadIdx.x >> 5;
    const long e_base = ((long)blockIdx.x * 8 + wave) * 16;
    if (e_base >= E) return;                  // wave-uniform (E % 128 == 0 here)
    const int* jrow = eidx;                   // source nodes
    const int* irow = eidx + E;               // target nodes (softmax segments)
    const int lm   = lane & 15;
    const int koff = (lane >> 4) * 2;

    // Preload the 16x64 fp32 A-tile (edge_attr rows) into registers: 32 VGPRs/lane.
    const float2* arow = (const float2*)(edge_attr + (size_t)(e_base + lm) * De);
    v2f areg[16];
    #pragma unroll
    for (int ks = 0; ks < 16; ++ks) {
        float2 af = arow[(ks * 4 + koff) >> 1];
        areg[ks].x = af.x; areg[ks].y = af.y;
    }

    // Node indices for this lane's 8 output rows (M = mrow0 + v).
    const int mrow0 = (lane >> 4) * 8;
    int gi[8], gj[8];
    #pragma unroll
    for (int v = 0; v < 8; ++v) {
        long e = e_base + mrow0 + v;
        gi[v] = irow[e];
        gj[v] = jrow[e];
    }

    float acc[8] = {0.f, 0.f, 0.f, 0.f, 0.f, 0.f, 0.f, 0.f};
    #pragma unroll
    for (int t = 0; t < 4; ++t) {             // four 16-wide slices of H=64
        v8f c = {};
        const float* bcol = sWeT + t * 16 + lm;
        #pragma unroll
        for (int ks = 0; ks < 16; ++ks) {     // K = 64 in steps of 4
            int k = ks * 4 + koff;
            v2f b;
            b.x = bcol[(k)     * H];          // LDS reads, conflict-free across lm
            b.y = bcol[(k + 1) * H];
            c = __builtin_amdgcn_wmma_f32_16x16x4_f32(false, areg[ks], false, b,
                                                      (short)0, c, false, false);
        }
        const int   h   = t * 16 + lm;
        const float wsc = sWsc[h];
        #pragma unroll
        for (int v = 0; v < 8; ++v) {
            // h-column h of edge (e_base + mrow0 + v): add L2-resident gathers, tanh, score
            float val = c[v] + hcat[(size_t)gi[v] * HO + h]
                             + hcat[(size_t)gj[v] * HO + H + h];
            acc[v] += tanhf(val) * wsc;
        }
    }
    // Reduce over the 16 lanes of each half-wave (butterfly keeps halves separate).
    #pragma unroll
    for (int v = 0; v < 8; ++v) {
        float s = acc[v];
        s += __shfl_xor(s, 1, 32);
        s += __shfl_xor(s, 2, 32);
        s += __shfl_xor(s, 4, 32);
        s += __shfl_xor(s, 8, 32);
        acc[v] = s;
    }
    if (lm == 0) {                            // lane 0 -> edges 0..7, lane 16 -> edges 8..15
        #pragma unroll
        for (int v = 0; v < 8; ++v) {
            long e = e_base + mrow0 + v;
            logits[e] = acc[v];
            atomicMax(&seg_max_u[gi[v]], enc_f(acc[v]));
        }
    }
}

// ---------- kernel 3: e = exp(logit - segmax); segment sums ----------
__global__ void edge_exp(const float* __restrict__ logits, const int* __restrict__ irow,
                         const unsigned* __restrict__ seg_max_u,
                         float* __restrict__ seg_sum, float* __restrict__ expv, int E) {
    long e = (long)blockIdx.x * blockDim.x + threadIdx.x;
    if (e >= E) return;
    int   seg = irow[e];
    float m   = dec_f(seg_max_u[seg]);
    float ex  = expf(logits[e] - m);
    expv[e]   = ex;
    atomicAdd(&seg_sum[seg], ex);
}

// ---------- kernel 4: alpha = e / s[seg] ----------
__global__ void edge_norm(const float* __restrict__ expv, const int* __restrict__ irow,
                          const float* __restrict__ seg_sum, float* __restrict__ alpha, int E) {
    long e = (long)blockIdx.x * blockDim.x + threadIdx.x;
    if (e < E) alpha[e] = expv[e] / seg_sum[irow[e]];
}

// ---------- host-side orchestration ----------
static inline size_t align256(size_t v) { return (v + 255) & ~(size_t)255; }

extern "C" void kernel_launch(void* const* d_in, const int* in_sizes, int n_in,
                              void* d_out, int out_size, void* d_ws, size_t ws_size,
                              hipStream_t stream) {
    // setup_inputs order: x, edge_index, edge_attr, W_i, W_j, W_e, w_score
    const float* x   = (const float*)d_in[0];
    const int*   ei  = (const int*)  d_in[1];
    const float* ea  = (const float*)d_in[2];
    const float* Wi  = (const float*)d_in[3];
    const float* Wj  = (const float*)d_in[4];
    const float* We  = (const float*)d_in[5];
    const float* wsc = (const float*)d_in[6];
    float* alpha = (float*)d_out;

    const int H  = in_sizes[6];            // 64
    const int C  = in_sizes[3] / H;        // 256
    const int De = in_sizes[5] / H;        // 64
    const int N  = in_sizes[0] / C;        // 100000
    const int E  = in_sizes[1] / 2;        // 1600000
    const int HO = 2 * H;                  // 128 (fused hi|hj table width)

    // workspace layout (~65 MB total)
    char*  ws    = (char*)d_ws;
    size_t off   = 0;
    float* hcat  = (float*)(ws + off); off += align256((size_t)N * HO * 4);
    float* Wcat  = (float*)(ws + off); off += align256((size_t)C * HO * 4);
    float* WeT   = (float*)(ws + off); off += align256((size_t)De * H * 4);
    float* logit = (float*)(ws + off); off += align256((size_t)E * 4);
    float* expv  = (float*)(ws + off); off += align256((size_t)E * 4);
    unsigned* smax = (unsigned*)(ws + off); off += align256((size_t)N * 4);
    float* ssum  = (float*)(ws + off); off += align256((size_t)N * 4);
    (void)ws_size;

    int prepN = C * HO;                    // 32768 >= De*H
    prep_weights<<<(prepN + 255) / 256, 256, 0, stream>>>(Wi, Wj, We, Wcat, WeT, C, H, De);
    init_seg<<<(N + 255) / 256, 256, 0, stream>>>(smax, ssum, N);

    node_proj<<<(N + 15) / 16, 256, 0, stream>>>(x, Wcat, hcat, N, C, HO);

    edge_logits<<<(E + 127) / 128, 256, 0, stream>>>(ea, WeT, hcat, ei, wsc,
                                                     logit, smax, E, De, H, HO);

    edge_exp<<<(E + 255) / 256, 256, 0, stream>>>(logit, ei + E, smax, ssum, expv, E);
    edge_norm<<<(E + 255) / 256, 256, 0, stream>>>(expv, ei + E, ssum, alpha, E);
}